// DirectionalDummyNodes_8237747274339
// MI455X (gfx1250) — compile-verified
//
#include <hip/hip_runtime.h>
#include <hip/hip_bf16.h>
#include <math.h>

typedef float v2f __attribute__((ext_vector_type(2)));
typedef float v8f __attribute__((ext_vector_type(8)));

#define NB_        32
#define DIST_      5.0f
#define BEPS_      1e-6f
#define INV_SQRT_F 0.17677669529663689f
#define SCALE1     (1.0f/1024.0f)          /* (1/F)*invsqrtF*invsqrtF */
#define SCALE2     (INV_SQRT_F/32.0f)      /* (1/F)*invsqrtF          */

/* workspace layout, float units */
enum : int {
  OFF_ZMIN  = 0,                 /* 32 orderable-uint keys  */
  OFF_ZMAX  = 32,                /* 32 keys                 */
  OFF_SMAX  = 64,                /* 64 keys (b,s)           */
  OFF_DENOM = 128,               /* 64 f                    */
  OFF_P     = 192,               /* 32*2*8*32 = 16384 f     */
  OFF_BC    = OFF_P   + 16384,   /* 32x48 combined B matrix */
  OFF_TGK   = OFF_BC  + 1536,    /* 16384 f                 */
  OFF_TGV   = OFF_TGK + 16384,   /* 16384 f                 */
  OFF_DYN   = OFF_TGV + 16384    /* scores N*2 | u N*8 | qgw N*32 */
};

__device__ __forceinline__ unsigned fkey(float f) {
  unsigned u = __float_as_uint(f);
  return (u & 0x80000000u) ? ~u : (u | 0x80000000u);
}
__device__ __forceinline__ float fdec(unsigned k) {
  return (k & 0x80000000u) ? __uint_as_float(k & 0x7fffffffu)
                           : __uint_as_float(~k);
}
__device__ __forceinline__ float silu_(float x) { return x / (1.0f + expf(-x)); }

/* ---- precompute small derived matrices + zero accumulators ---- */
__global__ void k_init(const float* __restrict__ init_dummy,
                       const float* __restrict__ W_q_dummy,
                       const float* __restrict__ W_dot,
                       const float* __restrict__ Wk_gen2,
                       const float* __restrict__ W_q_graph,
                       float* __restrict__ ws) {
  __shared__ float qd[32];
  __shared__ float qW[32];
  int tid = threadIdx.x;
  if (tid < 32) {                       /* qd = invsqrtF * init_dummy @ W_q_dummy */
    float a = 0.f;
    for (int e = 0; e < 32; ++e) a += init_dummy[e] * W_q_dummy[e*32 + tid];
    qd[tid] = a * INV_SQRT_F;
  }
  __syncthreads();
  if (tid < 32) {                       /* qW = qd @ W_dot */
    float a = 0.f;
    for (int f = 0; f < 32; ++f) a += qd[f] * W_dot[f*32 + tid];
    qW[tid] = a;
  }
  __syncthreads();
  float* Bc = ws + OFF_BC;              /* 32 x 48 row-major */
  { /* A_k[f,h] = sum_g Wk_gen2[h, f*32+g] * qW[g]  -> cols 32..39, pad 40..47 */
    int f = tid >> 3, h = tid & 7;
    float a = 0.f;
    for (int g = 0; g < 32; ++g) a += Wk_gen2[h*1024 + f*32 + g] * qW[g];
    Bc[f*48 + 32 + h] = a;
    Bc[f*48 + 40 + h] = 0.f;
  }
  for (int idx = tid; idx < 1024; idx += 256) {  /* W_qd_comb -> cols 0..31 */
    int f = idx >> 5, g = idx & 31;
    float a = 0.f;
    for (int e = 0; e < 32; ++e) a += W_q_graph[f*32 + e] * W_dot[e*32 + g];
    Bc[f*48 + g] = a * INV_SQRT_F;
  }
  for (int i = tid; i < 16384; i += 256) ws[OFF_P + i] = 0.f;
  unsigned* zmin = (unsigned*)(ws + OFF_ZMIN);
  unsigned* zmax = (unsigned*)(ws + OFF_ZMAX);
  unsigned* smax = (unsigned*)(ws + OFF_SMAX);
  if (tid < 32) { zmin[tid] = 0xFFFFFFFFu; zmax[tid] = 0u; }
  if (tid < 64) { smax[tid] = 0u; ws[OFF_DENOM + tid] = 0.f; }
}

/* ---- segment min/max of z via orderable-uint atomics ---- */
__global__ void k_minmax(const float* __restrict__ pos, const int* __restrict__ batch,
                         float* __restrict__ ws, int N) {
  int n = blockIdx.x * blockDim.x + threadIdx.x;
  if (n >= N) return;
  int b = batch[n];
  unsigned k = fkey(pos[n*3 + 2]);
  atomicMin((unsigned*)(ws + OFF_ZMIN) + b, k);
  atomicMax((unsigned*)(ws + OFF_ZMAX) + b, k);
}

/* ---- WMMA f32 GEMM: [qgW | u] = X(Nx32) @ Bc(32x48) ---- */
__global__ void k_gemm(const float* __restrict__ X, float* __restrict__ ws, int N) {
  __shared__ float sB[32*48];
  int tid = threadIdx.x;
  const float* Bc = ws + OFF_BC;
  for (int i = tid; i < 1536; i += 256) sB[i] = Bc[i];
  __syncthreads();
  int wid = tid >> 5, lane = tid & 31;
  int m = lane & 15, hi = lane >> 4;
  int ntiles = (N + 15) >> 4;
  int tile = blockIdx.x * 8 + wid;
  if (tile >= ntiles) return;           /* wave-uniform: EXEC stays full for WMMA */
  int row0 = tile * 16;
  int rl = row0 + m; if (rl > N - 1) rl = N - 1;
  v8f c0 = 0.0f, c1 = 0.0f, c2 = 0.0f;
#pragma unroll
  for (int kk = 0; kk < 8; ++kk) {
    int kb = kk*4 + hi*2;               /* A: lane holds A[m, kb], A[m, kb+1] */
    v2f a;
    a.x = X[rl*32 + kb];
    a.y = X[rl*32 + kb + 1];
    v2f b0, b1, b2;                     /* B: lane holds B[kb, col], B[kb+1, col] */
    b0.x = sB[kb*48 + m];       b0.y = sB[(kb+1)*48 + m];
    b1.x = sB[kb*48 + 16 + m];  b1.y = sB[(kb+1)*48 + 16 + m];
    b2.x = sB[kb*48 + 32 + m];  b2.y = sB[(kb+1)*48 + 32 + m];
    c0 = __builtin_amdgcn_wmma_f32_16x16x4_f32(false, a, false, b0, (short)0, c0, false, false);
    c1 = __builtin_amdgcn_wmma_f32_16x16x4_f32(false, a, false, b1, (short)0, c1, false, false);
    c2 = __builtin_amdgcn_wmma_f32_16x16x4_f32(false, a, false, b2, (short)0, c2, false, false);
  }
  float* qgw = ws + OFF_DYN + 10*N;
  float* u   = ws + OFF_DYN + 2*N;
#pragma unroll
  for (int r = 0; r < 8; ++r) {         /* D[r + 8*hi, lane%16] */
    int row = row0 + r + 8*hi;
    if (row < N) {
      qgw[row*32 + m]      = c0[r];
      qgw[row*32 + 16 + m] = c1[r];
      if (m < 8) u[row*8 + m] = c2[r];
    }
  }
}

/* ---- phase 1a: scores + segment max ---- */
__global__ void k_p1a(const float* __restrict__ pos, const int* __restrict__ batch,
                      const float* __restrict__ bw, const float* __restrict__ Wk1,
                      float* __restrict__ ws, int N) {
  int n = blockIdx.x * blockDim.x + threadIdx.x;
  if (n >= N) return;
  int b = batch[n];
  float z = pos[n*3 + 2];
  float zmin = fdec(((unsigned*)(ws + OFF_ZMIN))[b]);
  float zmax = fdec(((unsigned*)(ws + OFF_ZMAX))[b]);
  float bot = z - zmin + DIST_, top = zmax + DIST_ - z;
  float ib = 1.f/(bot + BEPS_), it = 1.f/(top + BEPS_);
  float e0[8], e1[8];
#pragma unroll
  for (int j = 0; j < 8; ++j) {
    float w = bw[j];
    e0[j] = sinf(w*bot)*ib;
    e1[j] = sinf(w*top)*it;
  }
  const float* up = ws + OFF_DYN + 2*N + n*8;
  float s0 = 0.f, s1 = 0.f;
#pragma unroll
  for (int h = 0; h < 8; ++h) {
    float a0 = 0.f, a1 = 0.f;
#pragma unroll
    for (int j = 0; j < 8; ++j) { float w = Wk1[j*8 + h]; a0 += e0[j]*w; a1 += e1[j]*w; }
    float uh = up[h];
    s0 += silu_(a0)*uh;
    s1 += silu_(a1)*uh;
  }
  s0 *= SCALE1; s1 *= SCALE1;
  float* sc = ws + OFF_DYN;
  sc[n*2]   = s0;
  sc[n*2+1] = s1;
  atomicMax((unsigned*)(ws + OFF_SMAX) + b*2,     fkey(s0));
  atomicMax((unsigned*)(ws + OFF_SMAX) + b*2 + 1, fkey(s1));
}

/* ---- phase 1b: exp + segment sum ---- */
__global__ void k_p1b(const int* __restrict__ batch, float* __restrict__ ws, int N) {
  int n = blockIdx.x * blockDim.x + threadIdx.x;
  if (n >= N) return;
  int b = batch[n];
  float* sc = ws + OFF_DYN;
  float m0 = fdec(((unsigned*)(ws + OFF_SMAX))[b*2]);
  float m1 = fdec(((unsigned*)(ws + OFF_SMAX))[b*2 + 1]);
  float e0 = expf(sc[n*2]   - m0);
  float e1 = expf(sc[n*2+1] - m1);
  sc[n*2]   = e0;
  sc[n*2+1] = e1;
  atomicAdd(ws + OFF_DENOM + b*2,     e0);
  atomicAdd(ws + OFF_DENOM + b*2 + 1, e1);
}

/* ---- phase 1c: P[b,s,h,f] += attn * hv_h * x_f  (LDS-accumulated) ---- */
__global__ void k_p1c(const float* __restrict__ pos, const int* __restrict__ batch,
                      const float* __restrict__ X, const float* __restrict__ bw,
                      const float* __restrict__ Wv1, float* __restrict__ ws, int N) {
  __shared__ float Pl[4*512];
  int tid = threadIdx.x;
  for (int i = tid; i < 2048; i += 256) Pl[i] = 0.f;
  int start = blockIdx.x * 256;
  int bb = batch[start < N ? start : N - 1];
  __syncthreads();
  int n = start + tid;
  if (n < N) {
    int b = batch[n];
    float z = pos[n*3 + 2];
    float zmin = fdec(((unsigned*)(ws + OFF_ZMIN))[b]);
    float zmax = fdec(((unsigned*)(ws + OFF_ZMAX))[b]);
    float bot = z - zmin + DIST_, top = zmax + DIST_ - z;
    float ib = 1.f/(bot + BEPS_), it = 1.f/(top + BEPS_);
    float e0[8], e1[8];
#pragma unroll
    for (int j = 0; j < 8; ++j) {
      float w = bw[j];
      e0[j] = sinf(w*bot)*ib;
      e1[j] = sinf(w*top)*it;
    }
    float hv0[8], hv1[8];
#pragma unroll
    for (int h = 0; h < 8; ++h) {
      float a0 = 0.f, a1 = 0.f;
#pragma unroll
      for (int j = 0; j < 8; ++j) { float w = Wv1[j*8 + h]; a0 += e0[j]*w; a1 += e1[j]*w; }
      hv0[h] = silu_(a0);
      hv1[h] = silu_(a1);
    }
    const float* sc = ws + OFF_DYN;
    float a0 = sc[n*2]   / ws[OFF_DENOM + b*2];
    float a1 = sc[n*2+1] / ws[OFF_DENOM + b*2 + 1];
    float x[32];
#pragma unroll
    for (int f = 0; f < 32; ++f) x[f] = X[n*32 + f];
    int lb = b - bb;
    if (lb >= 0 && lb < 4) {
      float* base = Pl + lb*512;
#pragma unroll
      for (int h = 0; h < 8; ++h) {
        float c0 = a0*hv0[h], c1 = a1*hv1[h];
        float* p0 = base + h*32;
        float* p1 = base + 256 + h*32;
        for (int f = 0; f < 32; ++f) {
          atomicAdd(&p0[f], c0*x[f]);
          atomicAdd(&p1[f], c1*x[f]);
        }
      }
    } else {
      float* base = ws + OFF_P + b*512;
#pragma unroll
      for (int h = 0; h < 8; ++h) {
        float c0 = a0*hv0[h], c1 = a1*hv1[h];
        for (int f = 0; f < 32; ++f) {
          atomicAdd(base + h*32 + f,       c0*x[f]);
          atomicAdd(base + 256 + h*32 + f, c1*x[f]);
        }
      }
    }
  }
  __syncthreads();
  for (int i = tid; i < 2048; i += 256) {
    int lb = i >> 9;
    int b2 = bb + lb;
    float v = Pl[i];
    if (b2 < NB_ && v != 0.f) atomicAdd(ws + OFF_P + b2*512 + (i & 511), v);
  }
}

/* ---- phase 2 prep: dummy_out + Tg_k/Tg_v tables ---- */
__global__ void k_prep2(const float* __restrict__ init_dummy,
                        const float* __restrict__ Wv2,
                        const float* __restrict__ Wkg2,
                        const float* __restrict__ Wvg2,
                        float* __restrict__ ws,
                        float* __restrict__ dummy_out) {
  __shared__ float dsh[2048];
  int tid = threadIdx.x;
  for (int idx = tid; idx < 2048; idx += 256) {
    int b = idx >> 6, s = (idx >> 5) & 1, g = idx & 31;
    const float* Pb = ws + OFF_P + (b*2 + s)*256;
    float acc = 0.f;
    for (int h = 0; h < 8; ++h)
      for (int f = 0; f < 32; ++f)
        acc += Pb[h*32 + f] * Wv2[h*1024 + f*32 + g];
    float v = init_dummy[g] + acc * INV_SQRT_F;
    dsh[idx] = v;
    dummy_out[idx] = v;
  }
  __syncthreads();
  for (int idx = tid; idx < 16384; idx += 256) {
    int b = idx >> 9, s = (idx >> 8) & 1, h = (idx >> 5) & 7, g = idx & 31;
    const float* d = dsh + (b*2 + s)*32;
    float ak = 0.f, av = 0.f;
    for (int f = 0; f < 32; ++f) {
      float dv = d[f];
      ak += dv * Wkg2[h*1024 + f*32 + g];
      av += dv * Wvg2[h*1024 + f*32 + g];
    }
    ws[OFF_TGK + idx] = ak * INV_SQRT_F;
    ws[OFF_TGV + idx] = av * INV_SQRT_F;
  }
}

/* ---- phase 2: per-node graph attention + residual ---- */
__global__ void k_p2(const float* __restrict__ pos, const int* __restrict__ batch,
                     const float* __restrict__ X, const float* __restrict__ bw,
                     const float* __restrict__ Wkg1, const float* __restrict__ Wvg1,
                     float* __restrict__ ws, float* __restrict__ out, int N) {
  int n = blockIdx.x * blockDim.x + threadIdx.x;
  if (n >= N) return;
  int b = batch[n];
  float z = pos[n*3 + 2];
  float zmin = fdec(((unsigned*)(ws + OFF_ZMIN))[b]);
  float zmax = fdec(((unsigned*)(ws + OFF_ZMAX))[b]);
  float bot = z - zmin + DIST_, top = zmax + DIST_ - z;
  float ib = 1.f/(bot + BEPS_), it = 1.f/(top + BEPS_);
  float e0[8], e1[8];
#pragma unroll
  for (int j = 0; j < 8; ++j) {
    float w = bw[j];
    e0[j] = sinf(w*bot)*ib;
    e1[j] = sinf(w*top)*it;
  }
  float hkg0[8], hkg1[8], hvg0[8], hvg1[8];
#pragma unroll
  for (int h = 0; h < 8; ++h) {
    float a0 = 0.f, a1 = 0.f, c0 = 0.f, c1 = 0.f;
#pragma unroll
    for (int j = 0; j < 8; ++j) {
      float wk = Wkg1[j*8 + h], wv = Wvg1[j*8 + h];
      a0 += e0[j]*wk; a1 += e1[j]*wk;
      c0 += e0[j]*wv; c1 += e1[j]*wv;
    }
    hkg0[h] = silu_(a0); hkg1[h] = silu_(a1);
    hvg0[h] = silu_(c0); hvg1[h] = silu_(c1);
  }
  float q[32];
  const float* qp = ws + OFF_DYN + 10*N + n*32;
#pragma unroll
  for (int g = 0; g < 32; ++g) q[g] = qp[g];
  const float* Tgk = ws + OFF_TGK + b*512;
  const float* Tgv = ws + OFF_TGV + b*512;
  float sc0 = 0.f, sc1 = 0.f;
#pragma unroll
  for (int h = 0; h < 8; ++h) {
    const float* t0 = Tgk + h*32;
    const float* t1 = Tgk + 256 + h*32;
    float d0 = 0.f, d1 = 0.f;
    for (int g = 0; g < 32; ++g) { d0 += q[g]*t0[g]; d1 += q[g]*t1[g]; }
    sc0 += hkg0[h]*d0;
    sc1 += hkg1[h]*d1;
  }
  sc0 *= SCALE2; sc1 *= SCALE2;
  float mm = fmaxf(sc0, sc1);
  float ex0 = expf(sc0 - mm), ex1 = expf(sc1 - mm);
  float inv = 1.f/(ex0 + ex1);
  float a0 = ex0*inv, a1 = ex1*inv;
#pragma unroll
  for (int f = 0; f < 32; ++f) {
    float w0 = 0.f, w1 = 0.f;
#pragma unroll
    for (int h = 0; h < 8; ++h) {
      w0 += hvg0[h]*Tgv[h*32 + f];
      w1 += hvg1[h]*Tgv[256 + h*32 + f];
    }
    out[n*32 + f] = X[n*32 + f] + a0*w0 + a1*w1;
  }
}

extern "C" void kernel_launch(void* const* d_in, const int* in_sizes, int n_in,
                              void* d_out, int out_size, void* d_ws, size_t ws_size,
                              hipStream_t stream) {
  const float* pos  = (const float*)d_in[0];
  const int*   batch= (const int*)  d_in[1];
  const float* X    = (const float*)d_in[2];
  const float* bw   = (const float*)d_in[3];
  const float* Wqd  = (const float*)d_in[4];
  const float* Wk1  = (const float*)d_in[5];
  const float* Wk2  = (const float*)d_in[6];
  const float* Wv1  = (const float*)d_in[7];
  const float* Wv2  = (const float*)d_in[8];
  const float* Wqg  = (const float*)d_in[9];
  const float* Wkg1 = (const float*)d_in[10];
  const float* Wkg2 = (const float*)d_in[11];
  const float* Wvg1 = (const float*)d_in[12];
  const float* Wvg2 = (const float*)d_in[13];
  const float* Wdot = (const float*)d_in[14];
  const float* idum = (const float*)d_in[15];
  int N = in_sizes[1];
  float* ws  = (float*)d_ws;
  float* out = (float*)d_out;
  float* dummy_out = out + (size_t)N * 32;

  int nb = (N + 255) / 256;
  int ntiles = (N + 15) / 16;
  int gb = (ntiles + 7) / 8;

  k_init  <<<1, 256, 0, stream>>>(idum, Wqd, Wdot, Wk2, Wqg, ws);
  k_minmax<<<nb, 256, 0, stream>>>(pos, batch, ws, N);
  k_gemm  <<<gb, 256, 0, stream>>>(X, ws, N);
  k_p1a   <<<nb, 256, 0, stream>>>(pos, batch, bw, Wk1, ws, N);
  k_p1b   <<<nb, 256, 0, stream>>>(batch, ws, N);
  k_p1c   <<<nb, 256, 0, stream>>>(pos, batch, X, bw, Wv1, ws, N);
  k_prep2 <<<1, 256, 0, stream>>>(idum, Wv2, Wkg2, Wvg2, ws, dummy_out);
  k_p2    <<<nb, 256, 0, stream>>>(pos, batch, X, bw, Wkg1, Wvg1, ws, out, N);
  (void)n_in; (void)out_size; (void)ws_size;
}